// QATM_7937099563457
// MI455X (gfx1250) — compile-verified
//
#include <hip/hip_runtime.h>
#include <stdint.h>

// QATM confidence kernel for MI455X (gfx1250, wave32).
//
// out[b,r] = max_q sqrt(softmax_r(a*x)[b,r,q] * softmax_q(a*x)[b,r,q])
//          = 2^( max_q( cm*x[b,r,q] - 0.5*L2ref[b,q] ) - 0.5*L2qry[b,r] )
// where cr = a_ref*log2(e), cq = a_qry*log2(e), cm = 0.5*(cr+cq),
//       L2ref[b,q] = log2 sum_r 2^(cr*x),  L2qry[b,r] = log2 sum_q 2^(cq*x).
//
// x = 256 MB fp32; read exactly twice. Pass B runs in REVERSE block order so
// its first reads hit the ~192 MB tail of x still resident in L2 from pass A
// (LRU-friendly), cutting HBM traffic from ~512 MB toward ~330 MB (~15 us at
// 23.3 TB/s). Bandwidth-bound; no contraction structure -> WMMA inapplicable.
// CDNA5 paths: GLOBAL_LOAD_ASYNC_TO_LDS_B128 (ASYNCcnt) + s_wait_asynccnt,
// b128 global/LDS accesses, v_exp_f32 / v_log_f32.

constexpr int kB      = 16;
constexpr int kR      = 4096;   // Hr*Wr
constexpr int kQ      = 1024;   // Hq*Wq
constexpr int kSlices = 32;     // r-slices for column-pass occupancy
constexpr int kRowsPerSlice = kR / kSlices;  // 128
constexpr float kLog2e = 1.4426950408889634f;

__device__ __forceinline__ float fexp2(float v) { return __builtin_amdgcn_exp2f(v); }
__device__ __forceinline__ float flog2(float v) { return __builtin_amdgcn_logf(v); }  // base-2

// ---- async global->LDS staging (gfx1250 GLOBAL_LOAD_ASYNC_TO_LDS_B128) ----
typedef int v4i __attribute__((ext_vector_type(4)));
typedef __attribute__((address_space(1))) v4i v4i_global;
typedef __attribute__((address_space(3))) v4i v4i_lds;

__device__ __forceinline__ void async_ld_b128(const float* g, float* l) {
#if __has_builtin(__builtin_amdgcn_global_load_async_to_lds_b128)
  // Flat->AS casts via uintptr_t: global flat addr == as1 addr; a flat
  // pointer into __shared__ carries the LDS byte offset in its low 32 bits,
  // which is exactly the 32-bit as3 pointer value.
  __builtin_amdgcn_global_load_async_to_lds_b128(
      (v4i_global*)(uintptr_t)g,
      (v4i_lds*)(uintptr_t)l,
      /*imm offset*/0, /*cpol*/0);
#else
  *(float4*)l = *(const float4*)g;  // fallback: plain load + ds_store
#endif
}

__device__ __forceinline__ void async_join() {
#if __has_builtin(__builtin_amdgcn_global_load_async_to_lds_b128)
#if __has_builtin(__builtin_amdgcn_s_wait_asynccnt)
  __builtin_amdgcn_s_wait_asynccnt(0);
#else
  asm volatile("s_wait_asynccnt 0" ::: "memory");
#endif
#endif
}

// ---------------------------------------------------------------------------
// Pass A1: partial column sums  part[(b,slice),q] = sum_{r in slice} 2^(cr*x)
// Grid: kB*kSlices = 512 blocks x 256 threads. Each thread owns 4 adjacent
// columns (float4 / global_load_b128); a block iteration reads 4 full rows
// (16 KB) perfectly coalesced, 8 b128 loads in flight per thread.
// ---------------------------------------------------------------------------
__global__ __launch_bounds__(256) void qatm_colsum(const float* __restrict__ x,
                                                   const float* __restrict__ coef_ref,
                                                   float* __restrict__ part) {
  const int tid = threadIdx.x;
  const int sl  = blockIdx.x & (kSlices - 1);
  const int b   = blockIdx.x >> 5;
  const int q4  = tid * 4;
  const float cr = coef_ref[0] * kLog2e;

  const float* p = x + ((size_t)b * kR + (size_t)sl * kRowsPerSlice) * kQ + q4;

  float4 s0 = {0.f, 0.f, 0.f, 0.f};
  float4 s1 = {0.f, 0.f, 0.f, 0.f};
  float4 s2 = {0.f, 0.f, 0.f, 0.f};
  float4 s3 = {0.f, 0.f, 0.f, 0.f};
  #pragma unroll 2
  for (int i = 0; i < kRowsPerSlice; i += 4) {
    float4 v0 = *(const float4*)(p + (size_t)(i + 0) * kQ);
    float4 v1 = *(const float4*)(p + (size_t)(i + 1) * kQ);
    float4 v2 = *(const float4*)(p + (size_t)(i + 2) * kQ);
    float4 v3 = *(const float4*)(p + (size_t)(i + 3) * kQ);
    s0.x += fexp2(cr * v0.x); s0.y += fexp2(cr * v0.y);
    s0.z += fexp2(cr * v0.z); s0.w += fexp2(cr * v0.w);
    s1.x += fexp2(cr * v1.x); s1.y += fexp2(cr * v1.y);
    s1.z += fexp2(cr * v1.z); s1.w += fexp2(cr * v1.w);
    s2.x += fexp2(cr * v2.x); s2.y += fexp2(cr * v2.y);
    s2.z += fexp2(cr * v2.z); s2.w += fexp2(cr * v2.w);
    s3.x += fexp2(cr * v3.x); s3.y += fexp2(cr * v3.y);
    s3.z += fexp2(cr * v3.z); s3.w += fexp2(cr * v3.w);
  }
  float4 s;
  s.x = (s0.x + s1.x) + (s2.x + s3.x);
  s.y = (s0.y + s1.y) + (s2.y + s3.y);
  s.z = (s0.z + s1.z) + (s2.z + s3.z);
  s.w = (s0.w + s1.w) + (s2.w + s3.w);
  *(float4*)&part[((size_t)b * kSlices + sl) * kQ + q4] = s;
}

// ---------------------------------------------------------------------------
// Pass A2: combine slice partials -> lhalf[b,q] = 0.5*log2(sum)   (16K floats)
// ---------------------------------------------------------------------------
__global__ __launch_bounds__(256) void qatm_combine(const float* __restrict__ part,
                                                    float* __restrict__ lhalf) {
  const int idx = blockIdx.x * 256 + threadIdx.x;   // 0..kB*kQ-1
  const int b = idx >> 10;
  const int q = idx & (kQ - 1);
  float s = 0.f;
  #pragma unroll
  for (int sl = 0; sl < kSlices; ++sl)
    s += part[((size_t)b * kSlices + sl) * kQ + q];
  lhalf[idx] = 0.5f * flog2(s);
}

// ---------------------------------------------------------------------------
// Pass B: per row r, fuse row-LSE and final max.
// Grid: kB*(kR/8) = 8192 blocks x 256 threads (8 wave32s, one row per wave).
// Blocks are processed in REVERSE order so the second read of x starts on the
// L2-resident tail left by pass A. lhalf[b,*] (4 KB) is staged to LDS with
// async b128 copies, then each wave streams its 4 KB row with float4 loads.
// ---------------------------------------------------------------------------
__global__ __launch_bounds__(256) void qatm_rowpass(const float* __restrict__ x,
                                                    const float* __restrict__ coef_ref,
                                                    const float* __restrict__ coef_qry,
                                                    const float* __restrict__ lhalf,
                                                    float* __restrict__ out) {
  __shared__ float h[kQ];
  const int tid  = threadIdx.x;
  const int bid  = (int)gridDim.x - 1 - (int)blockIdx.x;  // reverse for L2 reuse
  const int b    = bid >> 9;              // 512 blocks per batch
  const int rblk = bid & 511;

  // stage 0.5*L2ref row-block into LDS: one async b128 per thread (4 KB total)
  async_ld_b128(lhalf + (size_t)b * kQ + tid * 4, &h[tid * 4]);
  async_join();
  __syncthreads();

  const float cr = coef_ref[0] * kLog2e;
  const float cq = coef_qry[0] * kLog2e;
  const float cm = 0.5f * (cr + cq);

  const int wave = tid >> 5;
  const int lane = tid & 31;
  const int r    = rblk * 8 + wave;
  const float* row = x + ((size_t)b * kR + r) * kQ;

  float s = 0.f;
  float m = -3.0e38f;
  #pragma unroll
  for (int i = 0; i < kQ; i += 128) {
    const int q4 = i + lane * 4;
    float4 v  = *(const float4*)(row + q4);    // global_load_b128
    float4 hh = *(const float4*)(&h[q4]);      // ds_load_b128
    s += fexp2(cq * v.x);  m = fmaxf(m, __builtin_fmaf(cm, v.x, -hh.x));
    s += fexp2(cq * v.y);  m = fmaxf(m, __builtin_fmaf(cm, v.y, -hh.y));
    s += fexp2(cq * v.z);  m = fmaxf(m, __builtin_fmaf(cm, v.z, -hh.z));
    s += fexp2(cq * v.w);  m = fmaxf(m, __builtin_fmaf(cm, v.w, -hh.w));
  }

  // wave32 cross-lane reductions
  #pragma unroll
  for (int off = 16; off > 0; off >>= 1) {
    s += __shfl_xor(s, off, 32);
    m  = fmaxf(m, __shfl_xor(m, off, 32));
  }

  if (lane == 0)
    out[(size_t)b * kR + r] = fexp2(m - 0.5f * flog2(s));
}

// ---------------------------------------------------------------------------
extern "C" void kernel_launch(void* const* d_in, const int* in_sizes, int n_in,
                              void* d_out, int out_size, void* d_ws, size_t ws_size,
                              hipStream_t stream) {
  const float* x        = (const float*)d_in[0];
  const float* coef_ref = (const float*)d_in[1];
  const float* coef_qry = (const float*)d_in[2];
  float* out = (float*)d_out;

  float* part  = (float*)d_ws;                         // kB*kSlices*kQ = 2 MB
  float* lhalf = part + (size_t)kB * kSlices * kQ;     // kB*kQ = 64 KB

  qatm_colsum <<<kB * kSlices,    256, 0, stream>>>(x, coef_ref, part);
  qatm_combine<<<(kB * kQ) / 256, 256, 0, stream>>>(part, lhalf);
  qatm_rowpass<<<kB * (kR / 8),   256, 0, stream>>>(x, coef_ref, coef_qry,
                                                    lhalf, out);
}